// AttnReducerRNN_2473901162587
// MI455X (gfx1250) — compile-verified
//
#include <hip/hip_runtime.h>
#include <math.h>

#define HDIM 1024
#define IDIM 1024
#define LDIM 16384
#define KCAT 2048
#define G3H  3072
#define EPSV 1e-8f

typedef __attribute__((ext_vector_type(2))) float v2f;
typedef __attribute__((ext_vector_type(8))) float v8f;

// ---------------------------------------------------------------------------
// One wave computes dot(W[rowBase+m, k0:k0+kn], x[k0:k0+kn]) for m = 0..15
// using V_WMMA_F32_16X16X4_F32 (full fp32 matrix op, CDNA5).
//   A (16x4 f32): lane L holds W[rowBase + (L&15)][k + 2*(L>>4) + {0,1}]
//   B (4x16 f32): column 0 = x chunk -> lanes 0/16 carry K=0,1 / K=2,3
//   C column 0:   lane 0 -> rows 0..7 in c[0..7]; lane 16 -> rows 8..15
// ---------------------------------------------------------------------------
__device__ __forceinline__ v8f wmma_rowdot16(const float* __restrict__ W, int ld,
                                             int rowBase, int k0, int kn,
                                             const float* __restrict__ sx) {
    const int lane = threadIdx.x & 31;
    const int m    = lane & 15;
    const int kh   = (lane >> 4) << 1;  // 0 or 2
    const float2* wrow2 = reinterpret_cast<const float2*>(
        W + (size_t)(rowBase + m) * (size_t)ld + (size_t)k0 + kh);
    const float2* xs2 = reinterpret_cast<const float2*>(sx + k0 + kh);
    v8f c = {};
#pragma unroll 4
    for (int k = 0; k < kn; k += 4) {
        float2 av = wrow2[k >> 1];
        float2 xv = xs2[k >> 1];
        v2f a; a.x = av.x; a.y = av.y;
        v2f b; b.x = (m == 0) ? xv.x : 0.0f; b.y = (m == 0) ? xv.y : 0.0f;
        c = __builtin_amdgcn_wmma_f32_16x16x4_f32(false, a, false, b,
                                                  (short)0, c, false, false);
    }
    return c;
}

// lanes 0 and 16 hold C column 0 -> store 16 contiguous dots
__device__ __forceinline__ void store_col0(float* dst, int rowBase, const v8f& c) {
    const int lane = threadIdx.x & 31;
    if ((lane & 15) == 0) {
        const int off = rowBase + ((lane >> 4) << 3);
#pragma unroll
        for (int j = 0; j < 8; ++j) dst[off + j] = c[j];
    }
}

// ---------------- stage 1: q partials = attn_w @ [emb_row; h0] --------------
// 64 row-tiles x 4 K-chunks = 256 waves = 32 blocks
__global__ void k1_qproj(const int* __restrict__ token, const float* __restrict__ hidden,
                         const float* __restrict__ emb, const float* __restrict__ attn_w,
                         float* __restrict__ part1) {
    __shared__ float sx[KCAT];
    const float* erow = emb + (size_t)token[0] * IDIM;
    for (int i = threadIdx.x; i < IDIM; i += blockDim.x) {
        sx[i] = erow[i];
        sx[IDIM + i] = hidden[i];
    }
    __syncthreads();
    const int gw   = blockIdx.x * 8 + (threadIdx.x >> 5);
    const int tile = gw >> 2;  // 0..63
    const int kc   = gw & 3;   // 0..3
    v8f c = wmma_rowdot16(attn_w, KCAT, tile * 16, kc * 512, 512, sx);
    store_col0(part1 + kc * HDIM, tile * 16, c);
}

// reduce q partials + bias, compute qn = max(||q||, eps). 1 block x 256.
__global__ void k1b_qfin(const float* __restrict__ part1, const float* __restrict__ attn_b,
                         float* __restrict__ q, float* __restrict__ qn) {
    __shared__ float red[256];
    float ss = 0.f;
    for (int i = threadIdx.x; i < HDIM; i += blockDim.x) {
        float v = part1[i] + part1[HDIM + i] + part1[2 * HDIM + i] +
                  part1[3 * HDIM + i] + attn_b[i];
        q[i] = v;
        ss = fmaf(v, v, ss);
    }
    red[threadIdx.x] = ss; __syncthreads();
    for (int s = 128; s > 0; s >>= 1) {
        if (threadIdx.x < s) red[threadIdx.x] += red[threadIdx.x + s];
        __syncthreads();
    }
    if (threadIdx.x == 0) qn[0] = fmaxf(sqrtf(red[0]), EPSV);
}

// ---------------- stage 2 (dominant 64MB): sims + row norms, fused ---------
// 1024 waves (16 rows each) = 128 blocks. WMMA does E.q, VALU co-executes |E|^2.
__global__ void k2_sims(const float* __restrict__ enc, const float* __restrict__ q,
                        const float* __restrict__ qn, float* __restrict__ sims) {
    __shared__ float sq[HDIM];
    __shared__ float sd[8][16];
    for (int i = threadIdx.x; i < HDIM; i += blockDim.x) sq[i] = q[i];
    __syncthreads();
    const int wave = threadIdx.x >> 5, lane = threadIdx.x & 31;
    const int m = lane & 15, kh = (lane >> 4) << 1;
    const int rowBase = (blockIdx.x * 8 + wave) * 16;
    const float2* wrow2 = reinterpret_cast<const float2*>(
        enc + (size_t)(rowBase + m) * HDIM + kh);
    const float2* xs2 = reinterpret_cast<const float2*>(sq + kh);
    v8f c = {};
    float nv = 0.f;
#pragma unroll 4
    for (int k = 0; k < HDIM; k += 4) {
        float2 av = wrow2[k >> 1];
        nv = fmaf(av.x, av.x, nv);
        nv = fmaf(av.y, av.y, nv);
        float2 xv = xs2[k >> 1];
        v2f a; a.x = av.x; a.y = av.y;
        v2f b; b.x = (m == 0) ? xv.x : 0.0f; b.y = (m == 0) ? xv.y : 0.0f;
        c = __builtin_amdgcn_wmma_f32_16x16x4_f32(false, a, false, b,
                                                  (short)0, c, false, false);
    }
    if ((lane & 15) == 0) {
        const int off = (lane >> 4) << 3;
#pragma unroll
        for (int j = 0; j < 8; ++j) sd[wave][off + j] = c[j];
    }
    nv += __shfl_xor(nv, 16, 32);  // lanes m and m+16 both hold row-m |E|^2
    __syncthreads();
    if (lane < 16) {
        float en = fmaxf(sqrtf(nv), EPSV);
        sims[rowBase + lane] = sd[wave][lane] / (qn[0] * en);
    }
}

// ---------------- stage 3: softmax over 16384 (1 block x 1024) -------------
__global__ void k3_softmax(const float* __restrict__ sims, float* __restrict__ attn) {
    __shared__ float red[1024];
    float v[16];
    const int base = threadIdx.x * 16;
    float mx = -3.402823466e38f;
#pragma unroll
    for (int i = 0; i < 16; ++i) { v[i] = sims[base + i]; mx = fmaxf(mx, v[i]); }
    red[threadIdx.x] = mx; __syncthreads();
    for (int s = 512; s > 0; s >>= 1) {
        if (threadIdx.x < s) red[threadIdx.x] = fmaxf(red[threadIdx.x], red[threadIdx.x + s]);
        __syncthreads();
    }
    mx = red[0]; __syncthreads();
    float sum = 0.f;
#pragma unroll
    for (int i = 0; i < 16; ++i) { v[i] = expf(v[i] - mx); sum += v[i]; }
    red[threadIdx.x] = sum; __syncthreads();
    for (int s = 512; s > 0; s >>= 1) {
        if (threadIdx.x < s) red[threadIdx.x] += red[threadIdx.x + s];
        __syncthreads();
    }
    const float inv = 1.0f / red[0];
#pragma unroll
    for (int i = 0; i < 16; ++i) attn[base + i] = v[i] * inv;
}

// ---------------- stage 4: ctx partials = w^T E (L2-resident pass) ---------
// block = (l-chunk of 512) x (h-slab of 256); 128 blocks, coalesced slabs
__global__ void k4_ctx(const float* __restrict__ enc, const float* __restrict__ attn,
                       float* __restrict__ part4) {
    __shared__ float sw[512];
    const int hb = blockIdx.x & 3;
    const int lc = blockIdx.x >> 2;
    const int l0 = lc * 512;
    for (int i = threadIdx.x; i < 512; i += blockDim.x) sw[i] = attn[l0 + i];
    __syncthreads();
    const int h = hb * 256 + threadIdx.x;
    const float* ecol = enc + h;
    float acc = 0.f;
#pragma unroll 4
    for (int l = 0; l < 512; ++l)
        acc = fmaf(sw[l], ecol[(size_t)(l0 + l) * HDIM], acc);
    part4[lc * HDIM + h] = acc;
}

__global__ void k4b_ctxsum(const float* __restrict__ part4, float* __restrict__ ctx) {
    const int h = blockIdx.x * 256 + threadIdx.x;
    float s = 0.f;
#pragma unroll
    for (int c = 0; c < 32; ++c) s += part4[c * HDIM + h];
    ctx[h] = s;
}

// ---------------- stage 5: x = relu(comb_w @ [emb_row; ctx] + b) -----------
__global__ void k5_comb(const int* __restrict__ token, const float* __restrict__ emb,
                        const float* __restrict__ ctx, const float* __restrict__ comb_w,
                        float* __restrict__ part5) {
    __shared__ float sx[KCAT];
    const float* erow = emb + (size_t)token[0] * IDIM;
    for (int i = threadIdx.x; i < IDIM; i += blockDim.x) {
        sx[i] = erow[i];
        sx[IDIM + i] = ctx[i];
    }
    __syncthreads();
    const int gw   = blockIdx.x * 8 + (threadIdx.x >> 5);
    const int tile = gw >> 2;
    const int kc   = gw & 3;
    v8f c = wmma_rowdot16(comb_w, KCAT, tile * 16, kc * 512, 512, sx);
    store_col0(part5 + kc * IDIM, tile * 16, c);
}

__global__ void k5b_relu(const float* __restrict__ part5, const float* __restrict__ comb_b,
                         float* __restrict__ xvec) {
    const int i = blockIdx.x * 256 + threadIdx.x;
    float v = part5[i] + part5[IDIM + i] + part5[2 * IDIM + i] +
              part5[3 * IDIM + i] + comb_b[i];
    xvec[i] = fmaxf(v, 0.f);
}

// ---------------- stage 6: GRU gate matvecs (w_ih @ x and w_hh @ h0) -------
// 384 row-tiles x 2 K-chunks = 768 waves = 96 blocks
__global__ void k6_gates(const float* __restrict__ xvec, const float* __restrict__ hidden,
                         const float* __restrict__ w_ih, const float* __restrict__ w_hh,
                         float* __restrict__ part6) {
    __shared__ float sx[2 * HDIM];
    for (int i = threadIdx.x; i < HDIM; i += blockDim.x) {
        sx[i] = xvec[i];
        sx[HDIM + i] = hidden[i];
    }
    __syncthreads();
    const int gw   = blockIdx.x * 8 + (threadIdx.x >> 5);
    const int tile = gw >> 1;  // 0..383
    const int kc   = gw & 1;   // 0..1
    const float* W;
    const float* xs;
    int wr, rowOut;
    if (tile < 192) { W = w_ih; xs = sx;        wr = tile * 16;         rowOut = wr; }
    else            { W = w_hh; xs = sx + HDIM; wr = (tile - 192) * 16; rowOut = G3H + wr; }
    v8f c = wmma_rowdot16(W, HDIM, wr, kc * 512, 512, xs);
    store_col0(part6 + kc * (2 * G3H), rowOut, c);
}

// gates + new hidden state (elementwise over 1024). 4 blocks x 256.
__global__ void k6b_hnew(const float* __restrict__ part6, const float* __restrict__ b_ih,
                         const float* __restrict__ b_hh, const float* __restrict__ hidden,
                         float* __restrict__ hnew, float* __restrict__ out_h) {
    const int j = blockIdx.x * 256 + threadIdx.x;
    const float* p0 = part6;
    const float* p1 = part6 + 2 * G3H;
    float gi_r = p0[j] + p1[j] + b_ih[j];
    float gh_r = p0[G3H + j] + p1[G3H + j] + b_hh[j];
    float gi_z = p0[HDIM + j] + p1[HDIM + j] + b_ih[HDIM + j];
    float gh_z = p0[G3H + HDIM + j] + p1[G3H + HDIM + j] + b_hh[HDIM + j];
    float gi_n = p0[2 * HDIM + j] + p1[2 * HDIM + j] + b_ih[2 * HDIM + j];
    float gh_n = p0[G3H + 2 * HDIM + j] + p1[G3H + 2 * HDIM + j] + b_hh[2 * HDIM + j];
    float r = 1.f / (1.f + expf(-(gi_r + gh_r)));
    float z = 1.f / (1.f + expf(-(gi_z + gh_z)));
    float n = tanhf(fmaf(r, gh_n, gi_n));
    float h = fmaf(1.f - z, n, z * hidden[j]);
    hnew[j] = h;
    out_h[j] = h;
}

// ---------------- stage 7: out = relu(h fc2^T + b) out_w^T + b. 1 block ----
__global__ void k7_out(const float* __restrict__ hnew, const float* __restrict__ fc2_w,
                       const float* __restrict__ fc2_b, const float* __restrict__ out_w,
                       const float* __restrict__ out_b, float* __restrict__ out0) {
    __shared__ float fc[64];
    const int wave = threadIdx.x >> 5, lane = threadIdx.x & 31;
    for (int r = 0; r < 8; ++r) {
        const int row = wave * 8 + r;
        const float* wr = fc2_w + (size_t)row * HDIM;
        float s = 0.f;
        for (int k = lane; k < HDIM; k += 32) s = fmaf(wr[k], hnew[k], s);
        for (int d = 16; d > 0; d >>= 1) s += __shfl_xor(s, d, 32);
        if (lane == 0) fc[row] = fmaxf(s + fc2_b[row], 0.f);
    }
    __syncthreads();
    if (wave == 0) {
        float s = fmaf(fc[lane], out_w[lane], fc[32 + lane] * out_w[32 + lane]);
        for (int d = 16; d > 0; d >>= 1) s += __shfl_xor(s, d, 32);
        if (lane == 0) out0[0] = s + out_b[0];
    }
}

extern "C" void kernel_launch(void* const* d_in, const int* in_sizes, int n_in,
                              void* d_out, int out_size, void* d_ws, size_t ws_size,
                              hipStream_t stream) {
    const int*   token  = (const int*)d_in[0];
    const float* hidden = (const float*)d_in[1];
    const float* enc    = (const float*)d_in[2];
    const float* emb    = (const float*)d_in[3];
    const float* attn_w = (const float*)d_in[4];
    const float* attn_b = (const float*)d_in[5];
    const float* comb_w = (const float*)d_in[6];
    const float* comb_b = (const float*)d_in[7];
    const float* w_ih   = (const float*)d_in[8];
    const float* w_hh   = (const float*)d_in[9];
    const float* b_ih   = (const float*)d_in[10];
    const float* b_hh   = (const float*)d_in[11];
    const float* fc2_w  = (const float*)d_in[12];
    const float* fc2_b  = (const float*)d_in[13];
    const float* out_w  = (const float*)d_in[14];
    const float* out_b  = (const float*)d_in[15];

    float* out = (float*)d_out;          // [ out(1) | h_new(1024) | attn(16384) ]
    float* f   = (float*)d_ws;
    float* part1 = f;                    //  4096
    float* q     = f + 4096;             //  1024
    float* qn    = f + 5120;             //     8
    float* sims  = f + 5128;             // 16384
    float* part4 = f + 21512;            // 32768
    float* ctx   = f + 54280;            //  1024
    float* part5 = f + 55304;            //  4096
    float* xvec  = f + 59400;            //  1024
    float* part6 = f + 60424;            // 12288
    float* hnew  = f + 72712;            //  1024
    float* attn  = out + 1 + HDIM;

    k1_qproj  <<<32,  256, 0, stream>>>(token, hidden, emb, attn_w, part1);
    k1b_qfin  <<<1,   256, 0, stream>>>(part1, attn_b, q, qn);
    k2_sims   <<<128, 256, 0, stream>>>(enc, q, qn, sims);
    k3_softmax<<<1,  1024, 0, stream>>>(sims, attn);
    k4_ctx    <<<128, 256, 0, stream>>>(enc, attn, part4);
    k4b_ctxsum<<<4,   256, 0, stream>>>(part4, ctx);
    k5_comb   <<<32,  256, 0, stream>>>(token, emb, ctx, comb_w, part5);
    k5b_relu  <<<4,   256, 0, stream>>>(part5, comb_b, xvec);
    k6_gates  <<<96,  256, 0, stream>>>(xvec, hidden, w_ih, w_hh, part6);
    k6b_hnew  <<<4,   256, 0, stream>>>(part6, b_ih, b_hh, hidden, hnew, out + 1);
    k7_out    <<<1,   256, 0, stream>>>(hnew, fc2_w, fc2_b, out_w, out_b, out);
}